// AutoEncoder_31327491457575
// MI455X (gfx1250) — compile-verified
//
#include <hip/hip_runtime.h>

#define N_NODES 20000
#define E_RAW   320000
#define E_TOT   (E_RAW + N_NODES)
#define HEADS   8
#define NEG_SLOPE 0.2f
#define BN_EPS  1e-5f

typedef __attribute__((ext_vector_type(2))) float v2f;
typedef __attribute__((ext_vector_type(8))) float v8f;

// ---------------------------------------------------------------------------
// GEMM: Y[M=20000, Nout] = X[M, K] * W[K, Nout], via V_WMMA_F32_16X16X4_F32.
// One wave per 16x16 tile; 4 waves per block cover 4 consecutive N-tiles.
// A frag (16x4): lanes 0-15 M=0..15 K={0,1}, lanes 16-31 M=0..15 K={2,3}.
// B frag (4x16): v0 = rows K=0 (lanes 0-15) / K=2 (lanes 16-31), v1 = K=1/K=3.
// C/D frag: VGPR r, lanes 0-15 -> (M=r, N=lane); lanes 16-31 -> (M=r+8).
// ---------------------------------------------------------------------------
__global__ __launch_bounds__(128) void gemm16x16_wmma(const float* __restrict__ X,
                                                      const float* __restrict__ W,
                                                      float* __restrict__ Y,
                                                      int K, int Nout) {
  const int lane = threadIdx.x;          // 0..31
  const int half = lane >> 4;            // 0 or 1
  const int l15  = lane & 15;
  const int tm   = blockIdx.x * 16;
  const int tn   = (blockIdx.y * 4 + threadIdx.y) * 16;
  const int arow = tm + l15;
  const int bcol = tn + l15;

  v8f acc = {};
  for (int k = 0; k < K; k += 4) {
    v2f a = *(const v2f*)(X + (size_t)arow * K + k + 2 * half);   // K+2*half, K+2*half+1
    v2f b;
    b.x = W[(size_t)(k + 2 * half + 0) * Nout + bcol];
    b.y = W[(size_t)(k + 2 * half + 1) * Nout + bcol];
    acc = __builtin_amdgcn_wmma_f32_16x16x4_f32(false, a, false, b,
                                                (short)0, acc, false, false);
  }
  float* yp = Y + (size_t)(tm + half * 8) * Nout + tn + l15;
  #pragma unroll
  for (int r = 0; r < 8; ++r)
    yp[(size_t)r * Nout] = acc[r];
}

// ---------------------------------------------------------------------------
// Attention scores: s_src[n,h] = sum_c H[n,h,c]*a_src[h,c]; same for dst.
// ---------------------------------------------------------------------------
__global__ void att_scores_k(const float* __restrict__ H,
                             const float* __restrict__ a_src,
                             const float* __restrict__ a_dst,
                             float* __restrict__ ssrc, float* __restrict__ sdst,
                             int C) {
  int i = blockIdx.x * blockDim.x + threadIdx.x;
  if (i >= N_NODES * HEADS) return;
  int n = i >> 3, h = i & 7;
  const float* hp = H + ((size_t)n * HEADS + h) * C;
  float ss = 0.f, sd = 0.f;
  for (int c = 0; c < C; ++c) {
    float v = hp[c];
    ss += v * a_src[h * C + c];
    sd += v * a_dst[h * C + c];
  }
  ssrc[i] = ss;
  sdst[i] = sd;
}

__global__ void fill_zero_k(unsigned* __restrict__ p, int n) {
  int i = blockIdx.x * blockDim.x + threadIdx.x;
  if (i < n) p[i] = 0u;
}

// ordered-uint encoding of floats so atomicMax(uint) == float max
__device__ __forceinline__ unsigned enc_f(float f) {
  unsigned u = __float_as_uint(f);
  return (u & 0x80000000u) ? ~u : (u | 0x80000000u);
}
__device__ __forceinline__ float dec_f(unsigned u) {
  return (u & 0x80000000u) ? __uint_as_float(u & 0x7FFFFFFFu)
                           : __uint_as_float(~u);
}

__device__ __forceinline__ void edge_sd(int e, const int* __restrict__ ei,
                                        int& s, int& d) {
  if (e < E_RAW) { s = ei[e]; d = ei[E_RAW + e]; }
  else           { s = e - E_RAW; d = s; }           // self-loops
}

// Pass 1: per-destination segment max of leaky_relu(s_src[src]+s_dst[dst])
__global__ void edge_max_k(const int* __restrict__ ei,
                           const float* __restrict__ ssrc,
                           const float* __restrict__ sdst,
                           unsigned* __restrict__ m_u) {
  int i = blockIdx.x * blockDim.x + threadIdx.x;
  if (i >= E_TOT * HEADS) return;
  int e = i >> 3, h = i & 7;
  int s, d; edge_sd(e, ei, s, d);
  float v = ssrc[s * HEADS + h] + sdst[d * HEADS + h];
  v = v > 0.f ? v : NEG_SLOPE * v;
  atomicMax(&m_u[d * HEADS + h], enc_f(v));
}

// Pass 2: ex = exp(e - m[dst]); denom[dst] += ex
__global__ void edge_exp_k(const int* __restrict__ ei,
                           const float* __restrict__ ssrc,
                           const float* __restrict__ sdst,
                           const unsigned* __restrict__ m_u,
                           float* __restrict__ ex, float* __restrict__ denom) {
  int i = blockIdx.x * blockDim.x + threadIdx.x;
  if (i >= E_TOT * HEADS) return;
  int e = i >> 3, h = i & 7;
  int s, d; edge_sd(e, ei, s, d);
  float v = ssrc[s * HEADS + h] + sdst[d * HEADS + h];
  v = v > 0.f ? v : NEG_SLOPE * v;
  float xv = expf(v - dec_f(m_u[d * HEADS + h]));
  ex[i] = xv;
  atomicAdd(&denom[d * HEADS + h], xv);
}

// Pass 3: agg[dst,h,c] += H[src,h,c] * ex / (denom[dst,h]+1e-16)
__global__ void edge_agg_k(const int* __restrict__ ei,
                           const float* __restrict__ H,
                           const float* __restrict__ ex,
                           const float* __restrict__ denom,
                           float* __restrict__ agg, int C) {
  long long i = (long long)blockIdx.x * blockDim.x + threadIdx.x;
  long long total = (long long)E_TOT * HEADS * C;
  if (i >= total) return;
  int c = (int)(i % C);
  long long t = i / C;
  int h = (int)(t & 7);
  int e = (int)(t >> 3);
  int s, d; edge_sd(e, ei, s, d);
  float alpha = ex[(size_t)e * HEADS + h] / (denom[d * HEADS + h] + 1e-16f);
  atomicAdd(&agg[((size_t)d * HEADS + h) * C + c],
            H[((size_t)s * HEADS + h) * C + c] * alpha);
}

__global__ void final_concat_k(const float* __restrict__ agg,
                               const float* __restrict__ b,
                               float* __restrict__ out, int F) {
  int i = blockIdx.x * blockDim.x + threadIdx.x;
  if (i >= N_NODES * F) return;
  out[i] = agg[i] + b[i % F];
}

__global__ void final_mean_k(const float* __restrict__ agg,
                             const float* __restrict__ b,
                             float* __restrict__ out, int C) {
  int i = blockIdx.x * blockDim.x + threadIdx.x;
  if (i >= N_NODES * C) return;
  int n = i / C, c = i % C;
  float s = 0.f;
  #pragma unroll
  for (int h = 0; h < HEADS; ++h)
    s += agg[((size_t)n * HEADS + h) * C + c];
  out[i] = s * 0.125f + b[c];
}

// BatchNorm statistics: one block per feature column
__global__ __launch_bounds__(256) void bn_stats_k(const float* __restrict__ X,
                                                  float* __restrict__ mu,
                                                  float* __restrict__ rs, int F) {
  int f = blockIdx.x;
  float s = 0.f, s2 = 0.f;
  for (int n = threadIdx.x; n < N_NODES; n += blockDim.x) {
    float v = X[(size_t)n * F + f];
    s += v; s2 += v * v;
  }
  __shared__ float sh[256], sh2[256];
  sh[threadIdx.x] = s; sh2[threadIdx.x] = s2;
  __syncthreads();
  for (int off = 128; off > 0; off >>= 1) {
    if (threadIdx.x < off) {
      sh[threadIdx.x]  += sh[threadIdx.x + off];
      sh2[threadIdx.x] += sh2[threadIdx.x + off];
    }
    __syncthreads();
  }
  if (threadIdx.x == 0) {
    float m   = sh[0] * (1.f / N_NODES);
    float var = sh2[0] * (1.f / N_NODES) - m * m;
    mu[f] = m;
    rs[f] = rsqrtf(var + BN_EPS);
  }
}

__global__ void bn_relu_k(float* __restrict__ X, const float* __restrict__ mu,
                          const float* __restrict__ rs,
                          const float* __restrict__ g,
                          const float* __restrict__ b, int F) {
  int i = blockIdx.x * blockDim.x + threadIdx.x;
  if (i >= N_NODES * F) return;
  int f = i % F;
  float v = (X[i] - mu[f]) * rs[f] * g[f] + b[f];
  X[i] = v > 0.f ? v : 0.f;
}

// ---------------------------------------------------------------------------
// One GAT layer
// ---------------------------------------------------------------------------
static void run_gat(const float* X, const int* ei, const float* W,
                    const float* a_s, const float* a_d, const float* bias,
                    float* out, bool concat, int Fin, int C,
                    float* bufH, float* ssrc, float* sdst, unsigned* m_u,
                    float* denom, float* ex, float* agg, hipStream_t stream) {
  const int F = HEADS * C;
  dim3 gg(N_NODES / 16, F / 64);
  dim3 gb(32, 4);
  gemm16x16_wmma<<<gg, gb, 0, stream>>>(X, W, bufH, Fin, F);

  const int nh = N_NODES * HEADS;
  att_scores_k<<<(nh + 255) / 256, 256, 0, stream>>>(bufH, a_s, a_d, ssrc, sdst, C);
  fill_zero_k<<<(nh + 255) / 256, 256, 0, stream>>>(m_u, nh);
  fill_zero_k<<<(nh + 255) / 256, 256, 0, stream>>>((unsigned*)denom, nh);

  const int eh = E_TOT * HEADS;
  edge_max_k<<<(eh + 255) / 256, 256, 0, stream>>>(ei, ssrc, sdst, m_u);
  edge_exp_k<<<(eh + 255) / 256, 256, 0, stream>>>(ei, ssrc, sdst, m_u, ex, denom);

  const int aggn = N_NODES * F;
  fill_zero_k<<<(aggn + 255) / 256, 256, 0, stream>>>((unsigned*)agg, aggn);
  long long tot = (long long)E_TOT * HEADS * C;
  int blocks = (int)((tot + 255) / 256);
  edge_agg_k<<<blocks, 256, 0, stream>>>(ei, bufH, ex, denom, agg, C);

  if (concat)
    final_concat_k<<<(aggn + 255) / 256, 256, 0, stream>>>(agg, bias, out, F);
  else
    final_mean_k<<<(N_NODES * C + 255) / 256, 256, 0, stream>>>(agg, bias, out, C);
}

extern "C" void kernel_launch(void* const* d_in, const int* in_sizes, int n_in,
                              void* d_out, int out_size, void* d_ws, size_t ws_size,
                              hipStream_t stream) {
  const float* x     = (const float*)d_in[0];
  const int*   ei    = (const int*)  d_in[1];
  const float* We1   = (const float*)d_in[2];
  const float* as_e1 = (const float*)d_in[3];
  const float* ad_e1 = (const float*)d_in[4];
  const float* b_e1  = (const float*)d_in[5];
  const float* bn1_g = (const float*)d_in[6];
  const float* bn1_b = (const float*)d_in[7];
  const float* We2   = (const float*)d_in[8];
  const float* as_e2 = (const float*)d_in[9];
  const float* ad_e2 = (const float*)d_in[10];
  const float* b_e2  = (const float*)d_in[11];
  const float* Wd1   = (const float*)d_in[12];
  const float* as_d1 = (const float*)d_in[13];
  const float* ad_d1 = (const float*)d_in[14];
  const float* b_d1  = (const float*)d_in[15];
  const float* bn2_g = (const float*)d_in[16];
  const float* bn2_b = (const float*)d_in[17];
  const float* Wd2   = (const float*)d_in[18];
  const float* as_d2 = (const float*)d_in[19];
  const float* ad_d2 = (const float*)d_in[20];
  const float* b_d2  = (const float*)d_in[21];
  float* out = (float*)d_out;

  float* ws = (float*)d_ws;
  size_t off = 0;
  auto alloc = [&](size_t n) { float* p = ws + off; off += n; return p; };
  float*    bufH  = alloc((size_t)N_NODES * 512);
  float*    agg   = alloc((size_t)N_NODES * 512);
  float*    out1  = alloc((size_t)N_NODES * 128);
  float*    encb  = alloc((size_t)N_NODES * 32);
  float*    out3  = alloc((size_t)N_NODES * 128);
  float*    ssrc  = alloc((size_t)N_NODES * HEADS);
  float*    sdst  = alloc((size_t)N_NODES * HEADS);
  unsigned* m_u   = (unsigned*)alloc((size_t)N_NODES * HEADS);
  float*    denom = alloc((size_t)N_NODES * HEADS);
  float*    ex    = alloc((size_t)E_TOT * HEADS);
  float*    mu    = alloc(128);
  float*    rs    = alloc(128);
  (void)in_sizes; (void)n_in; (void)out_size; (void)ws_size;

  // encoder conv1: 64 -> 8x16 concat, then BN+ReLU
  run_gat(x, ei, We1, as_e1, ad_e1, b_e1, out1, true, 64, 16,
          bufH, ssrc, sdst, m_u, denom, ex, agg, stream);
  bn_stats_k<<<128, 256, 0, stream>>>(out1, mu, rs, 128);
  bn_relu_k<<<(N_NODES * 128 + 255) / 256, 256, 0, stream>>>(out1, mu, rs, bn1_g, bn1_b, 128);

  // encoder conv2: 128 -> 8x32 mean
  run_gat(out1, ei, We2, as_e2, ad_e2, b_e2, encb, false, 128, 32,
          bufH, ssrc, sdst, m_u, denom, ex, agg, stream);

  // decoder conv1: 32 -> 8x16 concat, then BN+ReLU
  run_gat(encb, ei, Wd1, as_d1, ad_d1, b_d1, out3, true, 32, 16,
          bufH, ssrc, sdst, m_u, denom, ex, agg, stream);
  bn_stats_k<<<128, 256, 0, stream>>>(out3, mu, rs, 128);
  bn_relu_k<<<(N_NODES * 128 + 255) / 256, 256, 0, stream>>>(out3, mu, rs, bn2_g, bn2_b, 128);

  // decoder conv2: 128 -> 8x64 mean -> d_out
  run_gat(out3, ei, Wd2, as_d2, ad_d2, b_d2, out, false, 128, 64,
          bufH, ssrc, sdst, m_u, denom, ex, agg, stream);
}